// MemNN_53575422050613
// MI455X (gfx1250) — compile-verified
//
#include <hip/hip_runtime.h>
#include <hip/hip_bf16.h>

#define VOCAB 50000
#define EMBD  128
#define STORY 50
#define SENT  20
#define BS    256
#define HOPS  3

typedef __attribute__((ext_vector_type(2))) float v2f;
typedef __attribute__((ext_vector_type(8))) float v8f;

// ---------------------------------------------------------------------------
// Kernel 1: fused gather+reduce over sentence words.
// One block per (b, s); 128 threads = one embedding element each.
// Reads each of the 4 tables ONCE per (b,s,j) index and produces both the
// position-encoded sums (m_k, tables 0..2) and plain sums (c_k, tables 1..3).
// ---------------------------------------------------------------------------
__global__ __launch_bounds__(EMBD) void memnn_gather(
    const int* __restrict__ x, const float* __restrict__ A,
    const float* __restrict__ TA, const float* __restrict__ TC,
    float* __restrict__ m, float* __restrict__ c)
{
    const int bs = blockIdx.x;            // b*STORY + s
    const int s  = bs % STORY;
    const int e  = threadIdx.x;

    const float* A0 = A;
    const float* A1 = A + 1l * VOCAB * EMBD;
    const float* A2 = A + 2l * VOCAB * EMBD;
    const float* A3 = A + 3l * VOCAB * EMBD;

    float w0 = 0.f, w1 = 0.f, w2 = 0.f;   // pe-weighted sums
    float s1 = 0.f, s2 = 0.f, s3 = 0.f;   // plain sums
    const float ke = (float)(e + 1) / (float)EMBD;

    for (int j = 0; j < SENT; ++j) {
        const int wd = x[bs * SENT + j];
        const long r = (long)wd * EMBD + e;
        const float a0 = A0[r], a1 = A1[r], a2 = A2[r], a3 = A3[r];
        const float jj = (float)(j + 1) / (float)SENT;
        const float pe = 1.f - jj - ke * (1.f - 2.f * jj);
        w0 += pe * a0; w1 += pe * a1; w2 += pe * a2;
        s1 += a1;      s2 += a2;      s3 += a3;
    }

    const float ta = TA[s * EMBD + e];
    const float tc = TC[s * EMBD + e];
    const long  o  = (long)bs * EMBD + e;
    const long  H  = (long)BS * STORY * EMBD;
    m[0 * H + o] = w0 + ta;               // m for hop 0 (table 0)
    m[1 * H + o] = w1 + ta;               // hop 1 (table 1)
    m[2 * H + o] = w2 + ta;               // hop 2 (table 2)
    c[0 * H + o] = s1 + tc;               // c for hop 0 (table 1)
    c[1 * H + o] = s2 + tc;               // hop 1 (table 2)
    c[2 * H + o] = s3 + tc;               // hop 2 (table 3)
}

// ---------------------------------------------------------------------------
// Kernel 2: query embedding + 3 attention hops. One block per batch row,
// 128 threads (4 wave32s). Dots use per-wave shfl_xor reductions.
// ---------------------------------------------------------------------------
__global__ __launch_bounds__(EMBD) void memnn_hops(
    const int* __restrict__ q, const float* __restrict__ A0,
    const float* __restrict__ m, const float* __restrict__ c,
    float* __restrict__ u_out)
{
    __shared__ float u_lds[EMBD];
    __shared__ float dots[STORY];
    const int b    = blockIdx.x;
    const int tid  = threadIdx.x;
    const int lane = tid & 31;
    const int wv   = tid >> 5;

    // u = sum_j A0[q[b,j]]
    float u_e = 0.f;
    for (int j = 0; j < SENT; ++j) {
        const int wd = q[b * SENT + j];
        u_e += A0[(long)wd * EMBD + tid];
    }
    u_lds[tid] = u_e;
    __syncthreads();

    const long H = (long)BS * STORY * EMBD;
    for (int k = 0; k < HOPS; ++k) {
        const float* mk = m + k * H + (long)b * STORY * EMBD;
        const float* ck = c + k * H + (long)b * STORY * EMBD;

        // dots[s] = <m_k[b,s,:], u>
        for (int s = wv; s < STORY; s += 4) {
            float p = 0.f;
            for (int i = 0; i < 4; ++i) {
                const int e = lane + 32 * i;
                p += mk[s * EMBD + e] * u_lds[e];
            }
            for (int off = 16; off > 0; off >>= 1)
                p += __shfl_xor(p, off, 32);
            if (lane == 0) dots[s] = p;
        }
        __syncthreads();

        // softmax over 50 (computed redundantly by all 128 threads)
        float mx = -3.4e38f;
        for (int s = 0; s < STORY; ++s) mx = fmaxf(mx, dots[s]);
        float sum = 0.f;
        for (int s = 0; s < STORY; ++s) sum += __expf(dots[s] - mx);
        const float inv = 1.f / sum;

        // o_e = sum_s p[s] * c_k[b,s,e];  u += o
        float o = 0.f;
        for (int s = 0; s < STORY; ++s)
            o += __expf(dots[s] - mx) * inv * ck[s * EMBD + tid];
        const float un = u_lds[tid] + o;
        __syncthreads();
        u_lds[tid] = un;
        __syncthreads();
    }
    u_out[b * EMBD + tid] = u_lds[tid];
}

// ---------------------------------------------------------------------------
// Kernel 3: logits = U[256,128] @ A3[50000,128]^T via V_WMMA_F32_16X16X4_F32.
// Block = 8 waves, one 16-wide vocab tile; wave w covers M rows 32w..32w+31
// (two 16x16 accumulators sharing the B fragment). 32 k-steps of K=4.
// A-frag layout: lane&15 = M, (lane>>4)*2 = K base. B-frag mirrored with N.
// ---------------------------------------------------------------------------
__global__ __launch_bounds__(256) void memnn_gemm_wmma(
    const float* __restrict__ U, const float* __restrict__ A3,
    float* __restrict__ out)
{
    const int v0    = blockIdx.x * 16;     // vocab tile base
    const int tid   = threadIdx.x;
    const int lane  = tid & 31;
    const int wv    = tid >> 5;            // 0..7
    const int n     = lane & 15;
    const int khalf = (lane >> 4) << 1;    // 0 or 2
    const int m0    = wv * 32;             // two M tiles per wave

    v8f acc0 = {};
    v8f acc1 = {};
    const float* brow  = A3 + (long)(v0 + n) * EMBD + khalf;
    const float* arow0 = U  + (long)(m0 + n) * EMBD + khalf;
    const float* arow1 = arow0 + 16 * EMBD;

    for (int k = 0; k < EMBD; k += 4) {
        v2f bf; bf.x = brow[k];  bf.y = brow[k + 1];
        v2f a0; a0.x = arow0[k]; a0.y = arow0[k + 1];
        v2f a1; a1.x = arow1[k]; a1.y = arow1[k + 1];
        acc0 = __builtin_amdgcn_wmma_f32_16x16x4_f32(
            false, a0, false, bf, (short)0, acc0, false, false);
        acc1 = __builtin_amdgcn_wmma_f32_16x16x4_f32(
            false, a1, false, bf, (short)0, acc1, false, false);
    }

    // D layout: vgpr i -> M = i + 8*(lane>>4), N = lane&15
    const int col   = v0 + n;
    const int rbase = (lane >> 4) * 8;
    for (int i = 0; i < 8; ++i) {
        out[(long)(m0 + rbase + i)      * VOCAB + col] = acc0[i];
        out[(long)(m0 + 16 + rbase + i) * VOCAB + col] = acc1[i];
    }
}

// ---------------------------------------------------------------------------
// Kernel 4: per-row logsumexp (online max/sum merge), one block per batch row.
// ---------------------------------------------------------------------------
__global__ __launch_bounds__(256) void memnn_row_lse(
    const float* __restrict__ out, float* __restrict__ lse)
{
    __shared__ float smx[256], ssm[256];
    const int b   = blockIdx.x;
    const int tid = threadIdx.x;
    float mx = -3.4e38f, sm = 0.f;
    for (int v = tid; v < VOCAB; v += 256) {
        const float xv = out[(long)b * VOCAB + v];
        if (xv > mx) { sm = sm * __expf(mx - xv) + 1.f; mx = xv; }
        else         { sm += __expf(xv - mx); }
    }
    smx[tid] = mx; ssm[tid] = sm;
    __syncthreads();
    for (int off = 128; off > 0; off >>= 1) {
        if (tid < off) {
            const float m2 = smx[tid + off], s2 = ssm[tid + off];
            const float M  = fmaxf(smx[tid], m2);
            ssm[tid] = ssm[tid] * __expf(smx[tid] - M) + s2 * __expf(m2 - M);
            smx[tid] = M;
        }
        __syncthreads();
    }
    if (tid == 0) lse[b] = smx[0] + __logf(ssm[0]);
}

// ---------------------------------------------------------------------------
// Kernel 5: out = logits - lse[b]  (in-place log_softmax correction)
// ---------------------------------------------------------------------------
__global__ __launch_bounds__(256) void memnn_finalize(
    float* __restrict__ out, const float* __restrict__ lse)
{
    const long i = (long)blockIdx.x * blockDim.x + threadIdx.x;
    if (i < (long)BS * VOCAB) {
        const int b = (int)(i / VOCAB);
        out[i] -= lse[b];
    }
}

extern "C" void kernel_launch(void* const* d_in, const int* in_sizes, int n_in,
                              void* d_out, int out_size, void* d_ws, size_t ws_size,
                              hipStream_t stream)
{
    const int*   x  = (const int*)d_in[0];     // [256,50,20]
    const int*   q  = (const int*)d_in[1];     // [256,20]
    const float* A  = (const float*)d_in[2];   // [4,50000,128]
    const float* TA = (const float*)d_in[3];   // [50,128]
    const float* TC = (const float*)d_in[4];   // [50,128]
    float* out = (float*)d_out;                // [256,50000]

    // Workspace layout (floats): m[3][256][50][128], c[3][...], u[256][128], lse[256]
    // Total ~39.5 MB.
    float* ws  = (float*)d_ws;
    const long H = (long)BS * STORY * EMBD;
    float* m   = ws;
    float* c   = m + 3 * H;
    float* u   = c + 3 * H;
    float* lse = u + (long)BS * EMBD;

    memnn_gather  <<<BS * STORY,            EMBD, 0, stream>>>(x, A, TA, TC, m, c);
    memnn_hops    <<<BS,                    EMBD, 0, stream>>>(q, A, m, c, u);
    memnn_gemm_wmma<<<VOCAB / 16,           256,  0, stream>>>(u, A + 3l * VOCAB * EMBD, out);
    memnn_row_lse <<<BS,                    256,  0, stream>>>(out, lse);
    memnn_finalize<<<((long)BS * VOCAB + 255) / 256, 256, 0, stream>>>(out, lse);
}